// RNNClassifierChain_36919538876512
// MI455X (gfx1250) — compile-verified
//
#include <hip/hip_runtime.h>
#include <hip/hip_bf16.h>
#include <math.h>

#define BB     64
#define NBITS  4096
#define HH     64
#define IN_    4098
#define LMAX_  320
#define G3     192          // 3*H
#define NTILE  257          // ceil(4098/16)
#define MROWS  (BB * LMAX_) // 20480
#define NBLK   (MROWS / 16) // 1280

typedef float v2f __attribute__((ext_vector_type(2)));
typedef float v8f __attribute__((ext_vector_type(8)));

// ---------------------------------------------------------------------------
// Kernel 1: build token sequences. One wave per batch row; ballot-compaction
// produces the sorted positive-column list (== jnp.sort(where(aug>0, col, IN))).
// ---------------------------------------------------------------------------
__global__ void k_build_tokens(const float* __restrict__ true_fp,
                               const int*   __restrict__ lfs,
                               int* __restrict__ inp_t,
                               int* __restrict__ tgt) {
  __shared__ int posbuf[LMAX_ + 1];
  const int b    = blockIdx.x;
  const int lane = threadIdx.x;

  for (int i = lane; i < LMAX_ + 1; i += 32) posbuf[i] = 0;
  __syncthreads();

  int count = 0; // total positives in aug row (== lengths)
  for (int base = 0; base < IN_; base += 32) {
    int col = base + lane;
    bool pred = false;
    if (col < IN_) {
      if (col == 0 || col == IN_ - 1) pred = true;
      else pred = (true_fp[b * NBITS + lfs[col - 1]] > 0.0f);
    }
    unsigned mask = (unsigned)__ballot(pred);
    int off = count + __popc(mask & ((1u << lane) - 1u));
    if (pred && off <= LMAX_) posbuf[off] = col;
    count += __popc(mask);
  }
  __syncthreads();

  int stored = count < (LMAX_ + 1) ? count : (LMAX_ + 1);
  for (int t = lane; t < LMAX_; t += 32) {
    int lab = (t < stored) ? posbuf[t] : 0;
    inp_t[b * LMAX_ + t] = lab;
    int tg = (t + 1 < stored) ? posbuf[t + 1] : 0;
    tgt[b * LMAX_ + t] = (t < count - 1) ? tg : 0;
  }
}

// ---------------------------------------------------------------------------
// Kernel 2: GRU recurrence, single persistent workgroup (8 waves).
// gh = h @ W_hh via V_WMMA_F32_16X16X4_F32, gates elementwise, h in LDS.
// dynamic LDS: h (64x64) + gh (64x192) = 64 KB.
// ---------------------------------------------------------------------------
__global__ void k_gru(const float* __restrict__ embeds,
                      const int*   __restrict__ inp_t,
                      const float* __restrict__ W_ih,
                      const float* __restrict__ b_ih,
                      const float* __restrict__ W_hh,
                      const float* __restrict__ b_hh,
                      float* __restrict__ outs) {
  extern __shared__ float smem[];
  float* hS  = smem;            // [64][64]
  float* ghS = smem + BB * HH;  // [64][192]

  const int tid  = threadIdx.x;
  const int lane = tid & 31;
  const int wave = tid >> 5;
  const int mloc = lane & 15;
  const int khalf = (lane >> 4) * 2;  // lanes 0-15 -> K 0/1, lanes 16-31 -> K 2/3

  for (int i = tid; i < BB * HH; i += blockDim.x) hS[i] = embeds[i];
  __syncthreads();

  for (int t = 0; t < LMAX_; ++t) {
    // --- gh = h @ W_hh : 4 (M) x 12 (N) tiles of 16x16, K=64 as 16 wmma steps
    for (int tile = wave; tile < 48; tile += 8) {
      const int mi = tile & 3;
      const int nj = tile >> 2;
      const int m = mi * 16 + mloc;
      const int n = nj * 16 + mloc;
      v8f c = {};
#pragma unroll
      for (int kk = 0; kk < 16; ++kk) {
        const int k0 = kk * 4 + khalf;
        v2f a, bf;
        a[0]  = hS[m * HH + k0];
        a[1]  = hS[m * HH + k0 + 1];
        bf[0] = W_hh[k0 * G3 + n];
        bf[1] = W_hh[(k0 + 1) * G3 + n];
        c = __builtin_amdgcn_wmma_f32_16x16x4_f32(false, a, false, bf,
                                                  (short)0, c, false, false);
      }
      const int rowbase = mi * 16 + ((lane >> 4) << 3);
#pragma unroll
      for (int r = 0; r < 8; ++r)
        ghS[(rowbase + r) * G3 + n] = c[r];
    }
    __syncthreads();

    // --- gates + state update (each element owned by exactly one thread)
    for (int e = tid; e < BB * HH; e += blockDim.x) {
      const int b = e >> 6, j = e & 63;
      const int tok = inp_t[b * LMAX_ + t];
      const float* gi = W_ih + tok * G3;
      const float gir = gi[j]          + b_ih[j];
      const float giz = gi[HH + j]     + b_ih[HH + j];
      const float gin = gi[2 * HH + j] + b_ih[2 * HH + j];
      const float ghr = ghS[b * G3 + j]          + b_hh[j];
      const float ghz = ghS[b * G3 + HH + j]     + b_hh[HH + j];
      const float ghn = ghS[b * G3 + 2 * HH + j] + b_hh[2 * HH + j];
      const float r = 1.0f / (1.0f + expf(-(gir + ghr)));
      const float z = 1.0f / (1.0f + expf(-(giz + ghz)));
      const float nn = tanhf(gin + r * ghn);
      const float hnew = (1.0f - z) * nn + z * hS[e];
      hS[e] = hnew;
      outs[(t * BB + b) * HH + j] = hnew;
    }
    __syncthreads();
  }
}

// ---------------------------------------------------------------------------
// Kernel 3: output projection + streamed log-softmax NLL.
// One wave per 16 (b,t) rows; 257 N-tiles of 16 classes via WMMA f32;
// online (max, sumexp) per C register, merged across 16 lanes by shuffles.
// ---------------------------------------------------------------------------
__global__ void k_proj(const float* __restrict__ outs,
                       const int*   __restrict__ tgt,
                       const float* __restrict__ W_out,
                       const float* __restrict__ b_out,
                       float* __restrict__ nll_part,
                       float* __restrict__ cnt_part) {
  __shared__ float aS[16 * HH];
  __shared__ float rowmax[16], rowsum[16], rowtgt[16];

  const int lane  = threadIdx.x;
  const int mbase = blockIdx.x * 16;  // m = b*LMAX + t
  const int mloc  = lane & 15;
  const int khalf = (lane >> 4) * 2;

  for (int i = lane; i < 16 * HH; i += 32) {
    const int r = i >> 6, k = i & 63;
    const int m = mbase + r;
    const int b = m / LMAX_, t = m % LMAX_;
    aS[i] = outs[(t * BB + b) * HH + k];
  }
  __syncthreads();

  // preload 16 A fragments (K = 0..63)
  v2f afrag[16];
#pragma unroll
  for (int kk = 0; kk < 16; ++kk) {
    afrag[kk][0] = aS[mloc * HH + kk * 4 + khalf];
    afrag[kk][1] = aS[mloc * HH + kk * 4 + khalf + 1];
  }

  float rmax[8], rsum[8];
#pragma unroll
  for (int r = 0; r < 8; ++r) { rmax[r] = -INFINITY; rsum[r] = 0.0f; }

  for (int nb = 0; nb < NTILE; ++nb) {
    const int n = nb * 16 + mloc;
    const int nclamp = (n < IN_) ? n : 0;
    if (nb + 1 < NTILE)
      __builtin_prefetch(&W_out[khalf * IN_ + (nb + 1) * 16 + mloc], 0, 1);
    v8f c = {};
#pragma unroll
    for (int kk = 0; kk < 16; ++kk) {
      const int k0 = kk * 4 + khalf;
      v2f bf;
      bf[0] = W_out[k0 * IN_ + nclamp];
      bf[1] = W_out[(k0 + 1) * IN_ + nclamp];
      c = __builtin_amdgcn_wmma_f32_16x16x4_f32(false, afrag[kk], false, bf,
                                                (short)0, c, false, false);
    }
    if (n < IN_) {
      const float bo = b_out[n];
#pragma unroll
      for (int r = 0; r < 8; ++r) {
        const float v = c[r] + bo;
        const float m2 = fmaxf(rmax[r], v);
        rsum[r] = rsum[r] * expf(rmax[r] - m2) + expf(v - m2);
        rmax[r] = m2;
      }
    }
  }

  // merge (max,sum) across the 16 lanes of each half-wave
#pragma unroll
  for (int s = 1; s < 16; s <<= 1) {
#pragma unroll
    for (int r = 0; r < 8; ++r) {
      const float om = __shfl_xor(rmax[r], s, 32);
      const float os = __shfl_xor(rsum[r], s, 32);
      const float m2 = fmaxf(rmax[r], om);
      rsum[r] = rsum[r] * expf(rmax[r] - m2) + os * expf(om - m2);
      rmax[r] = m2;
    }
  }
  if ((lane & 15) == 0) {
    const int rb = (lane >> 4) * 8;
#pragma unroll
    for (int r = 0; r < 8; ++r) { rowmax[rb + r] = rmax[r]; rowsum[rb + r] = rsum[r]; }
  }

  // target logit: direct 64-MAC dot per row (lanes 0..15)
  if (lane < 16) {
    const int c = tgt[mbase + lane];
    float acc = b_out[c];
    for (int k = 0; k < HH; ++k)
      acc += aS[lane * HH + k] * W_out[k * IN_ + c];
    rowtgt[lane] = acc;
  }
  __syncthreads();

  if (lane == 0) {
    float sn = 0.0f, sc = 0.0f;
    for (int r = 0; r < 16; ++r) {
      const int c = tgt[mbase + r];
      if (c != 0) {
        sn += (logf(rowsum[r]) + rowmax[r]) - rowtgt[r];
        sc += 1.0f;
      }
    }
    nll_part[blockIdx.x] = sn;
    cnt_part[blockIdx.x] = sc;
  }
}

// ---------------------------------------------------------------------------
// Kernel 4: deterministic final reduction.
// ---------------------------------------------------------------------------
__global__ void k_reduce(const float* __restrict__ nll_part,
                         const float* __restrict__ cnt_part,
                         float* __restrict__ out, int nparts) {
  __shared__ float sn[256], sc[256];
  const int tid = threadIdx.x;
  float an = 0.0f, ac = 0.0f;
  for (int i = tid; i < nparts; i += 256) { an += nll_part[i]; ac += cnt_part[i]; }
  sn[tid] = an; sc[tid] = ac;
  __syncthreads();
  for (int s = 128; s > 0; s >>= 1) {
    if (tid < s) { sn[tid] += sn[tid + s]; sc[tid] += sc[tid + s]; }
    __syncthreads();
  }
  if (tid == 0) out[0] = sn[0] / sc[0];
}

// ---------------------------------------------------------------------------
extern "C" void kernel_launch(void* const* d_in, const int* in_sizes, int n_in,
                              void* d_out, int out_size, void* d_ws, size_t ws_size,
                              hipStream_t stream) {
  const float* embeds  = (const float*)d_in[0];
  const float* true_fp = (const float*)d_in[1];
  // d_in[2] cand_fp, d_in[3] batch_ptr, d_in[4] labels: unused by reference math
  const int*   lfs     = (const int*)d_in[5];
  const float* W_ih    = (const float*)d_in[6];
  const float* b_ih    = (const float*)d_in[7];
  const float* W_hh    = (const float*)d_in[8];
  const float* b_hh    = (const float*)d_in[9];
  const float* W_out   = (const float*)d_in[10];
  const float* b_out   = (const float*)d_in[11];

  int*   inp_t    = (int*)d_ws;
  int*   tgt      = inp_t + BB * LMAX_;
  float* outs     = (float*)(tgt + BB * LMAX_);
  float* nll_part = outs + (size_t)LMAX_ * BB * HH;
  float* cnt_part = nll_part + NBLK;

  k_build_tokens<<<BB, 32, 0, stream>>>(true_fp, lfs, inp_t, tgt);
  k_gru<<<1, 256, (BB * HH + BB * G3) * sizeof(float), stream>>>(
      embeds, inp_t, W_ih, b_ih, W_hh, b_hh, outs);
  k_proj<<<NBLK, 32, 0, stream>>>(outs, tgt, W_out, b_out, nll_part, cnt_part);
  k_reduce<<<1, 256, 0, stream>>>(nll_part, cnt_part, (float*)d_out, NBLK);
}